// CoverTreeLoss_47098611367990
// MI455X (gfx1250) — compile-verified
//
#include <hip/hip_runtime.h>
#include <hip/hip_bf16.h>
#include <math.h>

typedef __attribute__((ext_vector_type(2))) float v2f;
typedef __attribute__((ext_vector_type(8))) float v8f;

#define T_CLS 10000
#define HID   512
#define BATCH 4096

// ---------------------------------------------------------------------------
// Kernel 1: added_weights[t] = weights[t] + sum(weights[ancestor in segment t])
// segment_id is sorted (np.repeat of arange) -> binary search the range.
// One block (128 threads, float4 each = 512 floats) per leaf class.
// ---------------------------------------------------------------------------
__global__ void build_added_weights(const float* __restrict__ weights,
                                    const int* __restrict__ anc_idx,
                                    const int* __restrict__ seg_id,
                                    int n_anc,
                                    float* __restrict__ added) {
    const int t   = blockIdx.x;
    const int tid = threadIdx.x;   // 0..127

    // lower_bound(t)
    int lo = 0, hi = n_anc;
    while (lo < hi) { int mid = (lo + hi) >> 1; if (seg_id[mid] <  t) lo = mid + 1; else hi = mid; }
    const int start = lo;
    // upper_bound(t)
    hi = n_anc;
    while (lo < hi) { int mid = (lo + hi) >> 1; if (seg_id[mid] <= t) lo = mid + 1; else hi = mid; }
    const int end = lo;

    float4 acc = ((const float4*)(weights + (size_t)t * HID))[tid];
    for (int e = start; e < end; ++e) {
        const int a = anc_idx[e];
        float4 w = ((const float4*)(weights + (size_t)a * HID))[tid];
        acc.x += w.x; acc.y += w.y; acc.z += w.z; acc.w += w.w;
    }
    ((float4*)(added + (size_t)t * HID))[tid] = acc;
}

// ---------------------------------------------------------------------------
// Kernel 2: logits[b][t] = sum_h x[b][h] * added[t][h], via V_WMMA_F32_16X16X4.
// Wave tile = 32x64 (2x4 grid of 16x16 WMMA tiles) -> 64 accumulator VGPRs,
// no spilling. ISA A/B operand layout (32-bit elements):
//   lane<16  holds K=h+0,h+1  (v2f);  lane>=16 holds K=h+2,h+3.
// A rows index batch; B columns index class (B[k][n] = added[t0+n][h+k]).
// Loads use uniform SGPR base + 32-bit VGPR element offsets (GVS addressing).
// 8 waves/block; 128 m-groups x 157 n-groups = 20096 waves = 2512 blocks.
// ---------------------------------------------------------------------------
__global__ void __launch_bounds__(256, 1)
gemm_wmma_f32(const float* __restrict__ x,
              const float* __restrict__ w,
              float* __restrict__ logits) {
    const int lane      = threadIdx.x & 31;
    const int waveInBlk = threadIdx.x >> 5;
    const int waveId    = blockIdx.x * 8 + waveInBlk;

    const int NG = 157;                 // ceil(625 / 4) n-tile groups
    const int mg = waveId / NG;         // 0..127
    const int ng = waveId % NG;
    if (mg >= (BATCH / 32)) return;     // uniform per wave

    const int laneM = lane & 15;
    const int hOff  = (lane >> 4) * 2;  // 0 for K=0,1 ; 2 for K=2,3
    const int b0 = mg * 32;
    const int t0 = ng * 64;

    // 32-bit element offsets (x: 2M elems, w: 5.12M elems -- both < 2^31)
    int xoff[2];
    int woff[4];
#pragma unroll
    for (int i = 0; i < 2; ++i)
        xoff[i] = (b0 + i * 16 + laneM) * HID + hOff;
#pragma unroll
    for (int j = 0; j < 4; ++j) {
        int t = t0 + j * 16 + laneM;
        if (t > T_CLS - 1) t = T_CLS - 1;   // clamp: duplicate loads, masked at store
        woff[j] = t * HID + hOff;
    }

    v8f acc[2][4];
#pragma unroll
    for (int i = 0; i < 2; ++i)
#pragma unroll
        for (int j = 0; j < 4; ++j)
            acc[i][j] = (v8f){};

#pragma unroll 4
    for (int h = 0; h < HID; h += 4) {
        v2f a[2], b[4];
#pragma unroll
        for (int i = 0; i < 2; ++i) a[i] = *(const v2f*)(x + xoff[i] + h);
#pragma unroll
        for (int j = 0; j < 4; ++j) b[j] = *(const v2f*)(w + woff[j] + h);
#pragma unroll
        for (int i = 0; i < 2; ++i)
#pragma unroll
            for (int j = 0; j < 4; ++j)
                acc[i][j] = __builtin_amdgcn_wmma_f32_16x16x4_f32(
                    false, a[i], false, b[j], (short)0, acc[i][j], false, false);
    }

    // C/D layout: VGPR r, lanes 0-15 -> (M=r, N=lane); lanes 16-31 -> (M=r+8, N=lane-16)
    const int mrow = (lane >> 4) * 8;
    const int ncol = lane & 15;
#pragma unroll
    for (int j = 0; j < 4; ++j) {
        const int t = t0 + j * 16 + ncol;
        if (t >= T_CLS) continue;           // per-lane predication (stores only)
#pragma unroll
        for (int i = 0; i < 2; ++i) {
            const size_t rowBase = (size_t)(b0 + i * 16 + mrow);
#pragma unroll
            for (int r = 0; r < 8; ++r)
                logits[(rowBase + r) * (size_t)T_CLS + t] = acc[i][j][r];
        }
    }
}

// ---------------------------------------------------------------------------
// Kernel 3: per-row online log-sum-exp; rowloss[b] = lse(logits[b]) - logits[b][y[b]]
// ---------------------------------------------------------------------------
__global__ void row_lse_loss(const float* __restrict__ logits,
                             const int* __restrict__ y,
                             float* __restrict__ rowloss) {
    const int b   = blockIdx.x;
    const int tid = threadIdx.x;
    const float* row = logits + (size_t)b * T_CLS;

    float m = -3.0e38f, s = 0.0f;
    for (int t = tid; t < T_CLS; t += 256) {
        float v = row[t];
        if (v > m) { s *= __expf(m - v); m = v; }
        s += __expf(v - m);
    }
    __shared__ float sm[256], ss[256];
    sm[tid] = m; ss[tid] = s;
    __syncthreads();
    for (int off = 128; off > 0; off >>= 1) {
        if (tid < off) {
            float m2 = sm[tid + off], s2 = ss[tid + off];
            float M = fmaxf(sm[tid], m2);
            ss[tid] = ss[tid] * __expf(sm[tid] - M) + s2 * __expf(m2 - M);
            sm[tid] = M;
        }
        __syncthreads();
    }
    if (tid == 0) {
        float lse = sm[0] + __logf(ss[0]);
        rowloss[b] = lse - row[y[b]];
    }
}

// ---------------------------------------------------------------------------
// Kernel 4: d_out[0] = mean(rowloss)
// ---------------------------------------------------------------------------
__global__ void reduce_mean(const float* __restrict__ rowloss,
                            float* __restrict__ out) {
    __shared__ float sh[256];
    const int tid = threadIdx.x;
    float s = 0.0f;
    for (int i = tid; i < BATCH; i += 256) s += rowloss[i];
    sh[tid] = s;
    __syncthreads();
    for (int off = 128; off > 0; off >>= 1) {
        if (tid < off) sh[tid] += sh[tid + off];
        __syncthreads();
    }
    if (tid == 0) out[0] = sh[0] / (float)BATCH;
}

extern "C" void kernel_launch(void* const* d_in, const int* in_sizes, int n_in,
                              void* d_out, int out_size, void* d_ws, size_t ws_size,
                              hipStream_t stream) {
    const float* x       = (const float*)d_in[0];
    const int*   y       = (const int*)  d_in[1];
    const float* weights = (const float*)d_in[2];
    const int*   anc     = (const int*)  d_in[3];
    const int*   seg     = (const int*)  d_in[4];
    const int    n_anc   = in_sizes[3];

    float* out     = (float*)d_out;
    float* logits  = out + 1;                        // d_out = [loss, logits...]
    float* added   = (float*)d_ws;                   // 10000*512 floats = 20.48 MB
    float* rowloss = added + (size_t)T_CLS * HID;    // +16 KB

    build_added_weights<<<T_CLS, 128, 0, stream>>>(weights, anc, seg, n_anc, added);
    gemm_wmma_f32<<<2512, 256, 0, stream>>>(x, added, logits);
    row_lse_loss<<<BATCH, 256, 0, stream>>>(logits, y, rowloss);
    reduce_mean<<<1, 256, 0, stream>>>(rowloss, out);
}